// RBFNetwork_45002667327625
// MI455X (gfx1250) — compile-verified
//
#include <hip/hip_runtime.h>
#include <hip/hip_bf16.h>

typedef float v2f __attribute__((ext_vector_type(2)));
typedef float v4f __attribute__((ext_vector_type(4)));
typedef float v8f __attribute__((ext_vector_type(8)));

constexpr int B_ = 2048, H_ = 512, F_ = 128, O_ = 10;

// ---- CDNA5 async global->LDS copy (ASYNCcnt path), with safe fallback ----
#if defined(__AMDGCN__) && __has_builtin(__builtin_amdgcn_global_load_async_to_lds_b128)
#define RBF_ASYNC_LDS 1
typedef int v4i __attribute__((vector_size(4 * sizeof(int))));
typedef __attribute__((address_space(1))) v4i gv4i;   // global (AS1)
typedef __attribute__((address_space(3))) v4i lv4i;   // LDS    (AS3)
#endif

__device__ __forceinline__ void cp16_global_to_lds(const float* g, float* l) {
#ifdef RBF_ASYNC_LDS
  __builtin_amdgcn_global_load_async_to_lds_b128(
      (gv4i*)g, (lv4i*)l, /*imm offset*/ 0, /*cpol*/ 0);
#else
  *(v4f*)l = *(const v4f*)g;   // sync fallback: global_load_b128 + ds_store_b128
#endif
}

__device__ __forceinline__ void wait_async_copies() {
#ifdef RBF_ASYNC_LDS
  asm volatile("s_wait_asynccnt 0x0" ::: "memory");
#endif
}

// ---------------------------------------------------------------------------
// Kernel 1: row squared-norms for inputs (2048 rows) and centers (512 rows).
// One wave32 per row; coalesced loads; shfl_xor tree reduction.
// ---------------------------------------------------------------------------
__global__ __launch_bounds__(256) void rbf_norms_kernel(
    const float* __restrict__ X, const float* __restrict__ C,
    float* __restrict__ xnorm, float* __restrict__ cnorm) {
  const int row  = blockIdx.x * 8 + (threadIdx.x >> 5);   // 320 blocks * 8 waves = 2560 rows
  const int lane = threadIdx.x & 31;
  const float* p = (row < B_) ? (X + row * F_) : (C + (row - B_) * F_);
  float s = 0.0f;
  #pragma unroll
  for (int j = 0; j < F_ / 32; ++j) {
    float v = p[lane + 32 * j];
    s += v * v;
  }
  #pragma unroll
  for (int off = 16; off > 0; off >>= 1) s += __shfl_xor(s, off, 32);
  if (lane == 0) {
    if (row < B_) xnorm[row] = s;
    else          cnorm[row - B_] = s;
  }
}

// ---------------------------------------------------------------------------
// Kernel 2: fused  S = X*C^T -> rbf = exp(-(|x|^2+|c|^2-2S)/(2s^2))
//                 -> out = rbf * W^T + b     (V_WMMA_F32_16X16X4_F32)
//
// 32 blocks x 256 threads (8 wave32). Block = 64 rows (4 M-tiles of 16).
// Wave w: M-tile = w&3, N-group q = w>>2. Each 4-wave q-group double-buffers
// 8KB center tiles into LDS with GLOBAL_LOAD_ASYNC_TO_LDS_B128 (ASYNCcnt),
// and all 4 waves consume each staged tile (ds_load_b64 B-fragments -> 32
// fp32 WMMAs). rbf tile is re-laid-out through LDS and folded into the
// second GEMM (4 WMMAs/tile). Cross-wave reduce via LDS ds_add_f32.
// Center L2 traffic: 8MB total (vs 32MB unshared).
// ---------------------------------------------------------------------------
__global__ __launch_bounds__(256) void rbf_fused_wmma_kernel(
    const float* __restrict__ X, const float* __restrict__ C,
    const float* __restrict__ sigma, const float* __restrict__ W,
    const float* __restrict__ bias, const float* __restrict__ xnorm,
    const float* __restrict__ cnorm, float* __restrict__ out) {
  __shared__ float cbuf[2][2][16 * F_];   // [buffer][q-group][16x128 tile] 32 KB
  __shared__ float stage[8][16][16];      // per-wave rbf staging, 8 KB
  __shared__ float outTile[4][16][16];    // per-M-tile accumulators, 4 KB

  const int tid   = threadIdx.x;
  const int wave  = tid >> 5;
  const int lane  = tid & 31;
  const int lhalf = lane >> 4;            // 0: lanes 0-15, 1: lanes 16-31
  const int lmod  = lane & 15;
  const int mq    = wave & 3;             // which M-tile of the block
  const int q     = wave >> 2;            // N-group: tiles q*16 .. q*16+15
  const int j     = tid & 127;            // thread index within q-group (128 thr)
  const int row_base = blockIdx.x * 64 + mq * 16;

  #pragma unroll
  for (int idx = tid; idx < 4 * 16 * 16; idx += 256)
    ((float*)outTile)[idx] = 0.0f;

  const float sg     = sigma[0];
  const float inv2s2 = 1.0f / (2.0f * sg * sg);

  // A fragments for this wave's 16 X-rows, whole K=128.
  // fp32 A 16x4 layout: lane -> row M=lane%16; VGPR0: K=2*lhalf, VGPR1: +1.
  v2f a[32];
  {
    const float* xrow = X + (size_t)(row_base + lmod) * F_;
    #pragma unroll
    for (int kk = 0; kk < 32; ++kk)
      a[kk] = *(const v2f*)(xrow + kk * 4 + 2 * lhalf);
  }

  float xnv[8];
  #pragma unroll
  for (int r = 0; r < 8; ++r) xnv[r] = xnorm[row_base + r + 8 * lhalf];

  // Prologue: async-stage tile i=0 of this q-group (contiguous 8KB block).
  {
    const float* gsrc = C + (size_t)(q * 16) * 16 * F_;
    float* ldst = &cbuf[0][q][0];
    #pragma unroll
    for (int c = 0; c < 4; ++c)
      cp16_global_to_lds(gsrc + (size_t)(j + 128 * c) * 4,
                         ldst + (size_t)(j + 128 * c) * 4);
  }
  wait_async_copies();
  __syncthreads();

  v8f out_acc = {};                       // [16 rows x 16 (O padded)] partial

  for (int i = 0; i < 16; ++i) {          // N-tiles of this q-group
    const int t        = q * 16 + i;
    const int col_base = t * 16;
    const int cur      = i & 1;

    // Kick off next tile's async copy into the other buffer.
    if (i + 1 < 16) {
      const float* gsrc = C + (size_t)(t + 1) * 16 * F_;
      float* ldst = &cbuf[1 - cur][q][0];
      #pragma unroll
      for (int c = 0; c < 4; ++c)
        cp16_global_to_lds(gsrc + (size_t)(j + 128 * c) * 4,
                           ldst + (size_t)(j + 128 * c) * 4);
    }

    // ---- GEMM 1: 32 fp32 WMMAs, B fragments from LDS (ds_load_b64) ----
    const float* cb = &cbuf[cur][q][lmod * F_];
    v8f acc = {};
    #pragma unroll
    for (int kc = 0; kc < 4; ++kc) {
      v2f bf[8];
      #pragma unroll
      for (int u = 0; u < 8; ++u)
        bf[u] = *(const v2f*)(cb + (kc * 8 + u) * 4 + 2 * lhalf);
      #pragma unroll
      for (int u = 0; u < 8; ++u)
        acc = __builtin_amdgcn_wmma_f32_16x16x4_f32(
            false, a[kc * 8 + u], false, bf[u], (short)0, acc, false, false);
    }

    // ---- distances -> exp, staged (D-layout -> row-major) ----
    const float cn = cnorm[col_base + lmod];
    #pragma unroll
    for (int r = 0; r < 8; ++r) {
      float dist = xnv[r] + cn - 2.0f * acc[r];
      stage[wave][r + 8 * lhalf][lmod] = __expf(-dist * inv2s2);
    }
    // stage[wave] is private to this wave: same-wave DS ordering suffices.

    // ---- GEMM 2: fold 16 H-values into out = rbf * W^T (O padded to 16) ----
    v2f a2[4], b2[4];
    #pragma unroll
    for (int c4 = 0; c4 < 4; ++c4) {
      const int kb = c4 * 4 + 2 * lhalf;                 // even -> b64 loads
      a2[c4] = *(const v2f*)&stage[wave][lmod][kb];
      const int o  = lmod;
      const int oc = (o < O_) ? o : (O_ - 1);            // branchless mask
      v2f wv = *(const v2f*)(W + (size_t)oc * H_ + col_base + kb);
      b2[c4].x = (o < O_) ? wv.x : 0.0f;
      b2[c4].y = (o < O_) ? wv.y : 0.0f;
    }
    #pragma unroll
    for (int c4 = 0; c4 < 4; ++c4)
      out_acc = __builtin_amdgcn_wmma_f32_16x16x4_f32(
          false, a2[c4], false, b2[c4], (short)0, out_acc, false, false);

    // Next tile's copy must have landed; all waves done with old buffer.
    wait_async_copies();
    __syncthreads();
  }

  // ---- reduce waves' partial output tiles (ds_add_f32 atomics) ----
  #pragma unroll
  for (int r = 0; r < 8; ++r)
    atomicAdd(&outTile[mq][r + 8 * lhalf][lmod], out_acc[r]);
  __syncthreads();

  // ---- bias + store the block's 64x10 outputs ----
  for (int idx = tid; idx < 64 * O_; idx += 256) {
    const int m = idx / O_, o = idx % O_;
    out[(size_t)(blockIdx.x * 64 + m) * O_ + o] =
        outTile[m >> 4][m & 15][o] + bias[o];
  }
}

// ---------------------------------------------------------------------------
extern "C" void kernel_launch(void* const* d_in, const int* in_sizes, int n_in,
                              void* d_out, int out_size, void* d_ws, size_t ws_size,
                              hipStream_t stream) {
  const float* X     = (const float*)d_in[0];  // [2048,128]
  const float* C     = (const float*)d_in[1];  // [512,128]
  const float* sigma = (const float*)d_in[2];  // [1]
  const float* W     = (const float*)d_in[3];  // [10,512]
  const float* bias  = (const float*)d_in[4];  // [10]
  float* out = (float*)d_out;                  // [2048,10]

  float* xnorm = (float*)d_ws;                 // 2048 floats
  float* cnorm = xnorm + B_;                   // 512 floats

  rbf_norms_kernel<<<(B_ + H_) / 8, 256, 0, stream>>>(X, C, xnorm, cnorm);
  rbf_fused_wmma_kernel<<<B_ / 64, 256, 0, stream>>>(
      X, C, sigma, W, bias, xnorm, cnorm, out);
}